// FourierBlock_39444979647103
// MI455X (gfx1250) — compile-verified
//
#include <hip/hip_runtime.h>

// FourierBlock on MI455X (gfx1250, wave32):
//   rfft(4096) via four-step FFT (64x64) using V_WMMA_F32_16X16X4_F32,
//   top-16 |X| selection, sparse 16-term inverse transform.
//   Strided (b,c)-column gather/scatter done by the Tensor Data Mover
//   (tensor_load_to_lds / tensor_store_from_lds, TENSORcnt-tracked).
// One 256-thread workgroup (8 waves) per (batch b, channel c) series.

#define TWO_PI 6.28318530717958647692f
#define TOPK   16

typedef __attribute__((ext_vector_type(2))) float v2f;
typedef __attribute__((ext_vector_type(8))) float v8f;

typedef __attribute__((ext_vector_type(4))) unsigned tdm_g0_t;
typedef __attribute__((ext_vector_type(8))) int      tdm_g1_t;
typedef __attribute__((ext_vector_type(4))) int      tdm_g2_t;

__device__ __forceinline__ v8f wmma_f32(v2f a, v2f b, v8f c) {
    // D = A(16x4 f32) * B(4x16 f32) + C(16x16 f32)
    return __builtin_amdgcn_wmma_f32_16x16x4_f32(
        /*neg_a=*/false, a, /*neg_b=*/false, b,
        /*c_mod=*/(short)0, c, /*reuse_a=*/false, /*reuse_b=*/false);
}

// Build a 2-D TDM descriptor for one strided column:
//   tile = 1 x 4096 elements of 4 bytes, row stride = 128 elements (the C dim).
__device__ __forceinline__ void tdm_make_desc(tdm_g0_t& g0, tdm_g1_t& g1,
                                              const void* lds_ptr, const void* gptr) {
    const unsigned lds_off = (unsigned)(size_t)lds_ptr;        // low dword of flat LDS addr
    const unsigned long long ga = (unsigned long long)(size_t)gptr;
    g0[0] = 1u;                                        // count=1, user descriptor
    g0[1] = lds_off;                                   // lds_addr (bytes)
    g0[2] = (unsigned)(ga & 0xffffffffu);              // global_addr[31:0]
    g0[3] = (unsigned)((ga >> 32) & 0x01ffffffu)       // global_addr[56:32]
          | 0x80000000u;                               // type=2 ("image")
    g1[0] = 0x00020000;   // workgroup_mask=0, data_size=2 (4 bytes), no flags
    g1[1] = 0x00010000;   // tensor_dim0 = 1          (bits 79:48, low 16 here)
    g1[2] = 0x10000000;   // tensor_dim1 = 4096       (bits 111:80, low 16 here)
    g1[3] = 0x00010000;   // tile_dim0   = 1          (bits 127:112)
    g1[4] = 0x00001000;   // tile_dim1   = 4096, tile_dim2 = 0
    g1[5] = 128;          // tensor_dim0_stride = 128 elements (bits 207:160, low 32)
    g1[6] = 0;            // stride0 hi, tensor_dim1_stride lo (unused for 2-D tile)
    g1[7] = 0;
}

__global__ __launch_bounds__(256)
void fourier_topk_kernel(const float* __restrict__ x, float* __restrict__ out) {
    constexpr int L  = 4096;   // sequence length
    constexpr int C  = 128;    // channels
    constexpr int NF = 2049;   // rfft bins

    const int c = blockIdx.x;          // channel
    const int b = blockIdx.y;          // batch
    const int tid   = threadIdx.x;
    const int lane  = tid & 31;
    const int wid   = tid >> 5;        // 8 waves
    const int l15   = lane & 15;
    const int lhalf = lane >> 4;       // 0 or 1

    // ---- LDS ----
    __shared__ float zr[64 * 65];      // Z real, padded rows (65) vs bank conflicts
    __shared__ float zi[64 * 65];      // Z imag; later reused as output staging
    __shared__ float xbuf[4160];       // phase 1: x matrix [64a+b]; phase 2: spec_r[0..2079], spec_i[2080..4159]
    __shared__ float c64t[64];         // cos(2*pi*i/64)
    __shared__ float wv[8];
    __shared__ int   wk[8];
    __shared__ int   sel_k[TOPK];
    __shared__ float sel_r[TOPK];
    __shared__ float sel_i[TOPK];
    __shared__ unsigned selmask[65];   // 2049-bit selected mask

    const float* xg = x   + ((size_t)b * L) * C + c;
    float*       og = out + ((size_t)b * L) * C + c;

    // ---- DMA the strided column into LDS: xbuf[n] = x[b, n, c] ----
    if (wid == 0) {
        tdm_g0_t g0; tdm_g1_t g1;
        tdm_make_desc(g0, g1, (const void*)xbuf, (const void*)xg);
        tdm_g2_t gz4 = {0, 0, 0, 0};
        tdm_g1_t gz8 = {0, 0, 0, 0, 0, 0, 0, 0};
        __builtin_amdgcn_tensor_load_to_lds(g0, g1, gz4, gz4, gz8, 0);
    }
    if (tid < 64) c64t[tid] = __cosf((float)tid * (TWO_PI / 64.0f));
    if (tid < 65) selmask[tid] = 0u;
    if (wid == 0) __builtin_amdgcn_s_wait_tensorcnt(0);
    __syncthreads();

    // ---- Step 1+2: Y = DFT64 * Xmat (two real matmuls), then twiddle -> Z ----
    // M = cf (64), K = a (64), N = bi (64): 16 output tiles, 2 per wave.
    // DFT64[m,k] = cos(2pi m k/64) - i sin(2pi m k/64):
    //   real part tile from c64t[idx], imag part tile from c64t[(idx+16)&63] (= -sin).
    for (int job = wid; job < 16; job += 8) {
        const int mt = job >> 2, nt = job & 3;
        v8f yr = {}; v8f yi = {};
        const int m = mt * 16 + l15;       // A row (cf)
        const int n = nt * 16 + l15;       // B col (bi)
        for (int ks = 0; ks < 16; ++ks) {
            const int k0 = ks * 4 + lhalf * 2;       // A layout: vgpr0 K=k0, vgpr1 K=k0+1
            const int p0 = (m * k0) & 63;
            const int p1 = (m * (k0 + 1)) & 63;
            v2f ac, as_, bx;
            ac.x  = c64t[p0];             ac.y  = c64t[p1];              //  cos
            as_.x = c64t[(p0 + 16) & 63]; as_.y = c64t[(p1 + 16) & 63];  // -sin
            bx.x = xbuf[k0 * 64 + n];     bx.y = xbuf[(k0 + 1) * 64 + n];
            yr = wmma_f32(ac,  bx, yr);
            yi = wmma_f32(as_, bx, yi);
        }
        // twiddle: Z = Y * exp(-2*pi*i*cf*bi/4096), store to LDS
        const int nn = nt * 16 + l15;                   // bi
        for (int j = 0; j < 8; ++j) {
            const int mm = mt * 16 + j + lhalf * 8;     // cf (C/D layout)
            const int p  = (mm * nn) & 4095;
            float sv, cv;
            __sincosf((float)p * (TWO_PI / 4096.0f), &sv, &cv);
            const float a = yr[j], bb = yi[j];
            zr[mm * 65 + nn] = a * cv + bb * sv;
            zi[mm * 65 + nn] = bb * cv - a * sv;
        }
    }
    __syncthreads();

    // ---- Step 3: X[cf+64d] = sum_bi E[d,bi] * Z[cf,bi], d = 0..31 (k <= 2047) ----
    // E = cos - i sin.  Xr = cos*Zr + sin*Zi ; Xi = cos*Zi - sin*Zr.
    // M = d (2 tiles), K = bi, N = cf (4 tiles): 8 jobs, exactly 1 per wave.
    {
        const int mt = wid >> 2;           // d tile 0..1
        const int nt = wid & 3;            // cf tile
        v8f xr = {}; v8f xj = {};
        const int m = mt * 16 + l15;       // d
        const int n = nt * 16 + l15;       // cf
        for (int ks = 0; ks < 16; ++ks) {
            const int k0 = ks * 4 + lhalf * 2;         // bi
            const int p0 = (m * k0) & 63;
            const int p1 = (m * (k0 + 1)) & 63;
            v2f ca, sa, nsa, br, bz;
            ca.x  = c64t[p0];             ca.y  = c64t[p1];              //  cos
            sa.x  = c64t[(p0 + 48) & 63]; sa.y  = c64t[(p1 + 48) & 63];  // +sin
            nsa.x = c64t[(p0 + 16) & 63]; nsa.y = c64t[(p1 + 16) & 63];  // -sin
            br.x = zr[n * 65 + k0];  br.y = zr[n * 65 + k0 + 1];  // Z^T read (padded rows)
            bz.x = zi[n * 65 + k0];  bz.y = zi[n * 65 + k0 + 1];
            xr = wmma_f32(ca,  br, xr);
            xr = wmma_f32(sa,  bz, xr);
            xj = wmma_f32(ca,  bz, xj);
            xj = wmma_f32(nsa, br, xj);
        }
        float* spec_r = xbuf;
        float* spec_i = xbuf + 2080;
        for (int j = 0; j < 8; ++j) {
            const int d  = mt * 16 + j + lhalf * 8;
            const int cf = nt * 16 + l15;
            const int k  = cf + 64 * d;            // 0..2047
            spec_r[k] = xr[j];
            spec_i[k] = xj[j];
        }
    }
    __syncthreads();

    // ---- Nyquist bin k = 2048: E[32,bi] = (-1)^bi ----
    if (tid == 0) {
        float sr = 0.f, si = 0.f;
        for (int t = 0; t < 64; ++t) {
            const float sg = (t & 1) ? -1.f : 1.f;
            sr += sg * zr[t];
            si += sg * zi[t];
        }
        xbuf[2048]        = sr;
        xbuf[2080 + 2048] = si;
    }
    __syncthreads();

    // ---- build 4096-entry cos table for the inverse (reuse zr region) ----
    float* ctab = zr;
    for (int j = 0; j < 16; ++j) {
        int i = tid + 256 * j;
        ctab[i] = __cosf((float)i * (TWO_PI / 4096.0f));
    }

    // ---- top-16 |X|^2 over k = 0..2048 (lowest-k tie break, like lax.top_k) ----
    const float* spec_r = xbuf;
    const float* spec_i = xbuf + 2080;
    for (int r = 0; r < TOPK; ++r) {
        float bv = -1.0f; int bk = 1 << 30;
        for (int j = 0; j < 9; ++j) {
            const int k = tid + 256 * j;
            if (k < NF && !((selmask[k >> 5] >> (k & 31)) & 1u)) {
                const float re = spec_r[k], im = spec_i[k];
                const float m2 = re * re + im * im;
                if (m2 > bv) { bv = m2; bk = k; }
            }
        }
        for (int off = 16; off > 0; off >>= 1) {
            const float ov = __shfl_down(bv, off, 32);
            const int   ok = __shfl_down(bk, off, 32);
            if (ov > bv || (ov == bv && ok < bk)) { bv = ov; bk = ok; }
        }
        if (lane == 0) { wv[wid] = bv; wk[wid] = bk; }
        __syncthreads();
        if (tid == 0) {
            float fv = wv[0]; int fk = wk[0];
            for (int w = 1; w < 8; ++w)
                if (wv[w] > fv || (wv[w] == fv && wk[w] < fk)) { fv = wv[w]; fk = wk[w]; }
            sel_k[r] = fk;
            selmask[fk >> 5] |= 1u << (fk & 31);
        }
        __syncthreads();
    }
    if (tid < TOPK) { const int k = sel_k[tid]; sel_r[tid] = spec_r[k]; sel_i[tid] = spec_i[k]; }
    __syncthreads();

    // ---- sparse inverse: 16 kept bins per series; stage result in LDS (zi) ----
    int   kk_[TOPK]; float rr_[TOPK]; float ii_[TOPK];
    #pragma unroll
    for (int t = 0; t < TOPK; ++t) { kk_[t] = sel_k[t]; rr_[t] = sel_r[t]; ii_[t] = sel_i[t]; }

    const float sc = 1.0f / 4096.0f;
    for (int j = 0; j < 16; ++j) {
        const int n = tid + 256 * j;
        float acc = 0.f;
        #pragma unroll
        for (int t = 0; t < TOPK; ++t) {
            const int k = kk_[t];
            if (k == 0) {
                acc += rr_[t];
            } else if (k == 2048) {
                acc += (n & 1) ? -rr_[t] : rr_[t];
            } else {
                const int p = (k * n) & 4095;
                const float cv = ctab[p];
                const float sv = ctab[(p + 3072) & 4095];   // sin = cos(th - pi/2)
                acc += 2.0f * (rr_[t] * cv - ii_[t] * sv);
            }
        }
        zi[n] = acc * sc;                 // stage output column in LDS
    }

    __syncthreads();
    // ---- DMA the output column LDS -> global (strided scatter) ----
    if (wid == 0) {
        tdm_g0_t g0; tdm_g1_t g1;
        tdm_make_desc(g0, g1, (const void*)zi, (const void*)og);
        tdm_g2_t gz4 = {0, 0, 0, 0};
        tdm_g1_t gz8 = {0, 0, 0, 0, 0, 0, 0, 0};
        __builtin_amdgcn_tensor_store_from_lds(g0, g1, gz4, gz4, gz8, 0);
        __builtin_amdgcn_s_wait_tensorcnt(0);   // LDS must stay live until DMA done
    }
}

extern "C" void kernel_launch(void* const* d_in, const int* in_sizes, int n_in,
                              void* d_out, int out_size, void* d_ws, size_t ws_size,
                              hipStream_t stream) {
    (void)in_sizes; (void)n_in; (void)d_ws; (void)ws_size; (void)out_size;
    const float* x  = (const float*)d_in[0];
    float* out      = (float*)d_out;
    dim3 grid(128, 64);   // (channel, batch): one WG per series
    fourier_topk_kernel<<<grid, dim3(256), 0, stream>>>(x, out);
}